// Net_60129542802
// MI455X (gfx1250) — compile-verified
//
#include <hip/hip_runtime.h>
#include <hip/hip_bf16.h>
#include <math.h>

// ---------------------------------------------------------------- constants
#define NN      8192
#define EE      262144
#define IN_CH   128
#define MP_CH   512
#define HID     256
#define KCLS    10
#define DELTA   0.311f
#define EPSV    0.001f
#define TV_CO   0.785f
#define BAL_CO  0.514f
#define QRANK   7372      // N - (N//K + 1) = 8192 - 820

typedef float v2f __attribute__((ext_vector_type(2)));
typedef float v8f __attribute__((ext_vector_type(8)));

// ---------------------------------------------------------------- GEMM (f32 WMMA)
// C[M,N] = A[M,K] @ B[K,N]  (+ optional bias & relu)
// block = 256 threads (8 waves), block tile = 128(M) x 64(N), K chunked by 64.
// Each wave owns a 16x64 strip: four 16x16 C tiles -> one A fragment feeds
// four V_WMMA_F32_16X16X4_F32. B is staged in *fragment order* (float2 pairs
// {B[4q+kb][c], B[4q+kb+2][c]}) so each B operand is one aligned ds_load_b64
// with no VGPR repacking in the hot loop.
#define TM 128
#define TN 64
#define TK 64

__global__ __launch_bounds__(256) void gemm_wmma_f32(
    const float* __restrict__ A, const float* __restrict__ B,
    const float* __restrict__ bias, float* __restrict__ C,
    int M, int N, int K, int act /*0=none, 1=bias+relu*/)
{
    __shared__ float lA[TM][TK + 4];            // 128x68 floats
    __shared__ v2f   lB2[(TK / 4) * 2 * TN];    // 2048 float2 = 16KB, fragment order

    const int tid  = threadIdx.x;
    const int lane = tid & 31;
    const int wave = tid >> 5;                 // 0..7 -> 16-row group
    const int rowB = blockIdx.x * TM;
    const int colB = blockIdx.y * TN;

    // WMMA fragment lane mapping (CDNA5 ISA 7.12.2):
    //  A 16x4 f32: lanes 0-15 -> M=lane, K={0,1}; lanes 16-31 -> M=lane-16, K={2,3}
    //  C 16x16 f32: VGPR i -> M=i (lanes 0-15) / M=8+i (lanes 16-31), N=lane&15
    const int mLoc  = wave * 16 + (lane & 15);
    const int khalf = (lane >> 4) * 2;         // 0 or 2
    const int nLoc  = lane & 15;
    const int kb    = lane >> 4;               // 0 or 1

    v8f acc[4] = {};

    for (int kk = 0; kk < K; kk += TK) {
        // stage A tile (128x64), coalesced along K
        #pragma unroll
        for (int i = tid; i < TM * TK; i += 256) {
            int r = i / TK, c = i % TK;
            lA[r][c] = A[(rowB + r) * K + kk + c];
        }
        // stage B tile (64x64) in fragment order: one float2 per thread-iter,
        // consecutive 8B LDS stores, coalesced global reads per (q,kb) row pair
        #pragma unroll
        for (int i2 = tid; i2 < (TK / 4) * 2 * TN; i2 += 256) {
            const int q  = i2 >> 7;           // k-quad 0..15
            const int hb = (i2 >> 6) & 1;     // kb 0..1
            const int c  = i2 & (TN - 1);     // col 0..63
            const int r0 = 4 * q + hb;
            v2f val;
            val.x = B[(kk + r0) * N + colB + c];
            val.y = B[(kk + r0 + 2) * N + colB + c];
            lB2[i2] = val;
        }
        __syncthreads();

        // prefetch next A chunk while this one computes (global_prefetch_b8)
        if (kk + TK < K)
            __builtin_prefetch(&A[(rowB + (tid >> 1)) * K + kk + TK + (tid & 1) * 32], 0, 0);

        #pragma unroll
        for (int k4 = 0; k4 < TK; k4 += 4) {
            const int q = k4 >> 2;
            v2f a;
            a.x = lA[mLoc][k4 + khalf];
            a.y = lA[mLoc][k4 + khalf + 1];
            const v2f* brow = &lB2[(q * 2 + kb) * TN];
            #pragma unroll
            for (int t = 0; t < 4; ++t) {
                const v2f b = brow[t * 16 + nLoc];
                acc[t] = __builtin_amdgcn_wmma_f32_16x16x4_f32(
                             false, a, false, b, (short)0, acc[t], false, false);
            }
        }
        __syncthreads();
    }

    const int row0 = rowB + wave * 16 + ((lane >> 4) ? 8 : 0);
    #pragma unroll
    for (int t = 0; t < 4; ++t) {
        const int col = colB + t * 16 + (lane & 15);
        const float bv = (act == 1) ? bias[col] : 0.0f;
        #pragma unroll
        for (int i = 0; i < 8; ++i) {
            float v = acc[t][i];
            if (act == 1) { v += bv; v = v > 0.0f ? v : 0.0f; }
            C[(row0 + i) * N + col] = v;
        }
    }
}

// ---------------------------------------------------------------- edge pass
// One wave32 per edge: L1 distance over 512 ch, gamma, scatter-add deg/agg.
__global__ __launch_bounds__(256) void gtv_edge(
    const float* __restrict__ h, const int* __restrict__ src,
    const int* __restrict__ dst, const float* __restrict__ w,
    float* __restrict__ deg, float* __restrict__ agg)
{
    const int e    = blockIdx.x * 8 + (threadIdx.x >> 5);
    const int lane = threadIdx.x & 31;
    if (e >= EE) return;

    const int si = src[e], di = dst[e];
    const float* hs = h + (size_t)si * MP_CH;
    const float* hd = h + (size_t)di * MP_CH;

    float vals[16];
    float partial = 0.0f;
    #pragma unroll
    for (int j = 0; j < 16; ++j) {
        const int c = lane + 32 * j;
        const float a = hs[c];
        vals[j] = a;
        partial += fabsf(a - hd[c]);
    }
    // wave32 butterfly reduce -> every lane holds the L1 distance
    #pragma unroll
    for (int off = 16; off > 0; off >>= 1)
        partial += __shfl_xor(partial, off, 32);

    const float gamma = w[e] / fmaxf(partial, EPSV);
    if (lane == 0) atomicAdd(&deg[di], gamma);
    float* ad = agg + (size_t)di * MP_CH;
    #pragma unroll
    for (int j = 0; j < 16; ++j)
        atomicAdd(&ad[lane + 32 * j], gamma * vals[j]);
}

// out = elu( h - delta*(deg*h - agg) + b )
__global__ void gtv_combine(const float* __restrict__ h, const float* __restrict__ deg,
                            const float* __restrict__ agg, const float* __restrict__ bias,
                            float* __restrict__ out)
{
    const int i = blockIdx.x * blockDim.x + threadIdx.x;
    if (i >= NN * MP_CH) return;
    const int row = i >> 9, col = i & (MP_CH - 1);
    const float hv = h[i];
    const float o  = hv - DELTA * (deg[row] * hv - agg[i]) + bias[col];
    out[i] = o > 0.0f ? o : (expf(o) - 1.0f);
}

// ---------------------------------------------------------------- head: logits + softmax
__global__ void logits_softmax(const float* __restrict__ z, const float* __restrict__ Wm2,
                               const float* __restrict__ bm2, float* __restrict__ logits,
                               float* __restrict__ s)
{
    const int row = blockIdx.x * blockDim.x + threadIdx.x;
    if (row >= NN) return;
    float acc[KCLS];
    #pragma unroll
    for (int k = 0; k < KCLS; ++k) acc[k] = bm2[k];
    const float* zr = z + (size_t)row * HID;
    for (int c = 0; c < HID; ++c) {
        const float zv = zr[c];
        const float* wr = Wm2 + c * KCLS;
        #pragma unroll
        for (int k = 0; k < KCLS; ++k) acc[k] += zv * wr[k];
    }
    float mx = acc[0];
    #pragma unroll
    for (int k = 1; k < KCLS; ++k) mx = fmaxf(mx, acc[k]);
    float ex[KCLS], sum = 0.0f;
    #pragma unroll
    for (int k = 0; k < KCLS; ++k) { ex[k] = expf(acc[k] - mx); sum += ex[k]; }
    const float inv = 1.0f / sum;
    #pragma unroll
    for (int k = 0; k < KCLS; ++k) {
        logits[row * KCLS + k] = acc[k];
        s[row * KCLS + k]      = ex[k] * inv;
    }
}

// ---------------------------------------------------------------- TV: edge sum + distinct-pair count
__global__ __launch_bounds__(256) void tv_edge(
    const float* __restrict__ s, const int* __restrict__ src,
    const int* __restrict__ dst, const float* __restrict__ w,
    unsigned* __restrict__ bitmap, float* __restrict__ tv_num,
    unsigned* __restrict__ edge_cnt)
{
    __shared__ float rs[256];
    __shared__ unsigned rc[256];
    const int e = blockIdx.x * blockDim.x + threadIdx.x;
    float t = 0.0f; unsigned c = 0;
    if (e < EE) {
        const int si = src[e], di = dst[e];
        float acc = 0.0f;
        #pragma unroll
        for (int k = 0; k < KCLS; ++k)
            acc += fabsf(s[si * KCLS + k] - s[di * KCLS + k]);
        t = w[e] * acc;
        const unsigned long long idx = (unsigned long long)si * (unsigned long long)NN + di;
        const unsigned word = (unsigned)(idx >> 5), bit = 1u << (idx & 31);
        const unsigned old = atomicOr(&bitmap[word], bit);
        if (!(old & bit)) c = 1;
    }
    rs[threadIdx.x] = t; rc[threadIdx.x] = c;
    __syncthreads();
    for (int off = 128; off > 0; off >>= 1) {
        if (threadIdx.x < (unsigned)off) {
            rs[threadIdx.x] += rs[threadIdx.x + off];
            rc[threadIdx.x] += rc[threadIdx.x + off];
        }
        __syncthreads();
    }
    if (threadIdx.x == 0) { atomicAdd(tv_num, rs[0]); atomicAdd(edge_cnt, rc[0]); }
}

// ---------------------------------------------------------------- per-column bitonic sort -> quantile
__global__ __launch_bounds__(1024) void col_sort_quant(const float* __restrict__ s,
                                                       float* __restrict__ quant)
{
    __shared__ float buf[NN];   // 32 KB of the 320 KB/WGP LDS
    const int k = blockIdx.x;
    for (int i = threadIdx.x; i < NN; i += 1024) buf[i] = s[i * KCLS + k];
    __syncthreads();
    for (int size = 2; size <= NN; size <<= 1) {
        for (int stride = size >> 1; stride > 0; stride >>= 1) {
            for (int i = threadIdx.x; i < NN; i += 1024) {
                const int j = i ^ stride;
                if (j > i) {
                    const bool up = ((i & size) == 0);
                    const float a = buf[i], b = buf[j];
                    if ((a > b) == up) { buf[i] = b; buf[j] = a; }
                }
            }
            __syncthreads();
        }
    }
    if (threadIdx.x == 0) quant[k] = buf[QRANK];
}

__global__ void asym_sum(const float* __restrict__ s, const float* __restrict__ quant,
                         float* __restrict__ out)
{
    __shared__ float red[256];
    const int i = blockIdx.x * blockDim.x + threadIdx.x;
    float v = 0.0f;
    if (i < NN * KCLS) {
        const int k = i % KCLS;
        const float t = s[i] - quant[k];
        v = (t >= 0.0f) ? (float)(KCLS - 1) * t : -t;
    }
    red[threadIdx.x] = v;
    __syncthreads();
    for (int off = 128; off > 0; off >>= 1) {
        if (threadIdx.x < (unsigned)off) red[threadIdx.x] += red[threadIdx.x + off];
        __syncthreads();
    }
    if (threadIdx.x == 0) atomicAdd(out, red[0]);
}

__global__ void finalize(const float* __restrict__ tv_num, const unsigned* __restrict__ cnt,
                         const float* __restrict__ asym, float* __restrict__ out2)
{
    const float n_edges = (float)(*cnt);
    out2[0] = TV_CO * (*tv_num) / (2.0f * n_edges);
    const float denom = (float)NN * (float)(KCLS - 1);
    out2[1] = BAL_CO * (denom - *asym) / denom;
}

// ---------------------------------------------------------------- launch
extern "C" void kernel_launch(void* const* d_in, const int* in_sizes, int n_in,
                              void* d_out, int out_size, void* d_ws, size_t ws_size,
                              hipStream_t stream)
{
    const float* x   = (const float*)d_in[0];
    const int*   ei  = (const int*)  d_in[1];
    const float* ew  = (const float*)d_in[2];
    const float* W1  = (const float*)d_in[3];
    const float* b1  = (const float*)d_in[4];
    const float* W2  = (const float*)d_in[5];
    const float* b2  = (const float*)d_in[6];
    const float* Wm1 = (const float*)d_in[7];
    const float* bm1 = (const float*)d_in[8];
    const float* Wm2 = (const float*)d_in[9];
    const float* bm2 = (const float*)d_in[10];
    const int* src = ei;
    const int* dst = ei + EE;

    float* out = (float*)d_out;                 // [81920 logits][tv][bal]

    // workspace layout (float offsets)
    float* ws   = (float*)d_ws;
    float* H    = ws;                            // 8192*512
    float* AGG  = ws + 4194304;                  // 8192*512
    float* H2   = ws + 8388608;                  // 8192*512
    float* Z    = ws + 12582912;                 // 8192*256
    float* S    = ws + 14680064;                 // 8192*10
    float* DEG  = ws + 14761984;                 // 8192
    float* QUANT= ws + 14770176;                 // 16
    float* SCAL = ws + 14770192;                 // [0]=tv_num [1]=asym
    unsigned* CNT    = (unsigned*)(ws + 14770194);
    unsigned* BITMAP = (unsigned*)(ws + 14770208); // 8192*8192/32 words = 8MB

    const dim3 blk256(256), blk1024(1024);
    const dim3 gEdge(EE / 8);                    // wave per edge, 8 waves/block
    const dim3 gComb((NN * MP_CH) / 256);

    // ---- layer 1
    hipMemsetAsync(AGG, 0, (size_t)NN * MP_CH * 4, stream);
    hipMemsetAsync(DEG, 0, (size_t)NN * 4, stream);
    gemm_wmma_f32<<<dim3(NN / TM, MP_CH / TN), blk256, 0, stream>>>(
        x, W1, nullptr, H, NN, MP_CH, IN_CH, 0);
    gtv_edge<<<gEdge, blk256, 0, stream>>>(H, src, dst, ew, DEG, AGG);
    gtv_combine<<<gComb, blk256, 0, stream>>>(H, DEG, AGG, b1, H2);

    // ---- layer 2
    hipMemsetAsync(AGG, 0, (size_t)NN * MP_CH * 4, stream);
    hipMemsetAsync(DEG, 0, (size_t)NN * 4, stream);
    gemm_wmma_f32<<<dim3(NN / TM, MP_CH / TN), blk256, 0, stream>>>(
        H2, W2, nullptr, H, NN, MP_CH, MP_CH, 0);
    gtv_edge<<<gEdge, blk256, 0, stream>>>(H, src, dst, ew, DEG, AGG);
    gtv_combine<<<gComb, blk256, 0, stream>>>(H, DEG, AGG, b2, H2);

    // ---- MLP head
    gemm_wmma_f32<<<dim3(NN / TM, HID / TN), blk256, 0, stream>>>(
        H2, Wm1, bm1, Z, NN, HID, MP_CH, 1);
    logits_softmax<<<dim3(NN / 256), blk256, 0, stream>>>(Z, Wm2, bm2, out, S);

    // ---- TV loss
    hipMemsetAsync(BITMAP, 0, (size_t)NN * NN / 8, stream);
    hipMemsetAsync(SCAL, 0, 16, stream);
    tv_edge<<<dim3(EE / 256), blk256, 0, stream>>>(S, src, dst, ew, BITMAP,
                                                   &SCAL[0], CNT);

    // ---- balance loss
    col_sort_quant<<<dim3(KCLS), blk1024, 0, stream>>>(S, QUANT);
    asym_sum<<<dim3((NN * KCLS) / 256), blk256, 0, stream>>>(S, QUANT, &SCAL[1]);

    finalize<<<dim3(1), dim3(1), 0, stream>>>(&SCAL[0], CNT, &SCAL[1],
                                              out + NN * KCLS);
}